// SpanDecoder_50302656971023
// MI455X (gfx1250) — compile-verified
//
#include <hip/hip_runtime.h>
#include <hip/hip_bf16.h>
#include <math.h>

// ---------------------------------------------------------------------------
// SpanDecoder for MI455X (gfx1250, wave32)
//
//   logits[t,s,:] , preds[t,s]  for t in [0,16384), s in [0,5)
//
// Decomposition:
//   gates(t,s) = G[t] + avg_contrib + T_prev[prev]          (all length-200)
//   G[t]       = enc[t] @ W_encT + (b_ih + b_hh)            <- WMMA GEMM
//   T_avg[j]   = label_emb[j] @ W_avgT    (9 x 200 table)
//   T_prev[j]  = label_emb[j] @ W_prevT   (9 x 200 table)
//   avg_contrib = T_avg[0]                    at t = 0
//               = mean_{6 idx} T_avg[idx]     afterwards (linearity of matvec)
// ---------------------------------------------------------------------------

typedef float v2f __attribute__((ext_vector_type(2)));
typedef float v8f __attribute__((ext_vector_type(8)));

#define SEQ_LEN   16384
#define ENC_DIM   768
#define EMB_DIM   50
#define GATES     200           // 4 * HID
#define N_LABELS  9
#define MAX_LEN   5
#define NT_TILES  13            // ceil(200 / 16)
#define NPAD      (NT_TILES * 16)   // 208
#define XDIM      868           // ENC_DIM + 2*EMB_DIM

__device__ __forceinline__ float sigmoidf_fast(float x) {
    return 1.0f / (1.0f + __expf(-x));
}

// ---------------------------------------------------------------------------
// Prep 1: repack W_enc = W_ih[:, 100:868] into k-pair-interleaved column-major:
//   wt2[(k/2)*NPAD + n] = { W_ih[n][100 + k], W_ih[n][100 + k + 1] }  (k even)
// so a B-fragment load is one coalesced b64 per lane.
// ---------------------------------------------------------------------------
__global__ void sd_prep_wt2(const float* __restrict__ Wih, float* __restrict__ wt2) {
    int idx = blockIdx.x * 256 + threadIdx.x;           // (k2, n) flattened
    if (idx >= (ENC_DIM / 2) * NPAD) return;
    int k2 = idx / NPAD;
    int n  = idx % NPAD;
    float lo = 0.0f, hi = 0.0f;
    if (n < GATES) {
        lo = Wih[n * XDIM + 100 + 2 * k2];
        hi = Wih[n * XDIM + 100 + 2 * k2 + 1];
    }
    wt2[idx * 2 + 0] = lo;
    wt2[idx * 2 + 1] = hi;
}

// ---------------------------------------------------------------------------
// Prep 2: bias sum + the two 9x200 embedding->gate tables.
// ---------------------------------------------------------------------------
__global__ void sd_prep_tables(const float* __restrict__ Wih,
                               const float* __restrict__ emb,
                               const float* __restrict__ bih,
                               const float* __restrict__ bhh,
                               float* __restrict__ bsum,
                               float* __restrict__ Tav,
                               float* __restrict__ Tpv) {
    int k = blockIdx.x * 256 + threadIdx.x;             // gate row 0..199
    if (k >= GATES) return;
    bsum[k] = bih[k] + bhh[k];
    for (int j = 0; j < N_LABELS; ++j) {
        float sa = 0.0f, sp = 0.0f;
        for (int e = 0; e < EMB_DIM; ++e) {
            float ev = emb[j * EMB_DIM + e];
            sa += ev * Wih[k * XDIM + e];
            sp += ev * Wih[k * XDIM + EMB_DIM + e];
        }
        Tav[j * GATES + k] = sa;
        Tpv[j * GATES + k] = sp;
    }
}

// ---------------------------------------------------------------------------
// GEMM: G[16384 x 200] = enc[16384 x 768] @ W_encT[768 x 200] + bsum
// One wave per 16-row stripe; A-stripe staged in LDS; 13 N-tiles, K in steps
// of 4 via V_WMMA_F32_16X16X4_F32 (fp32, matches reference precision).
//
// Fragment layouts (ISA 7.12.2):
//   A 16x4 f32 : lane L holds A[L&15][k0], A[L&15][k0+1] with k0 = (L>>4)*2
//   B 4x16 f32 : lane L holds B[k0][L&15], B[k0+1][L&15]
//   C/D 16x16  : VGPR r = rows r (lanes 0-15) and r+8 (lanes 16-31), N = lane&15
// ---------------------------------------------------------------------------
__global__ __launch_bounds__(32) void sd_gemm_gates(const float* __restrict__ enc,
                                                    const float* __restrict__ wt2,
                                                    const float* __restrict__ bsum,
                                                    float* __restrict__ G) {
    __shared__ float As[16 * ENC_DIM];                  // 48 KB A-stripe

    const int lane  = threadIdx.x;
    const int mbase = blockIdx.x * 16;

    // Coalesced cooperative load of the 16x768 stripe (float4 = b128 loads).
    const float4* encv = reinterpret_cast<const float4*>(enc + (size_t)mbase * ENC_DIM);
    float4*       asv  = reinterpret_cast<float4*>(As);
    #pragma unroll 4
    for (int i = 0; i < (16 * ENC_DIM / 4) / 32; ++i) {
        int idx = i * 32 + lane;
        asv[idx] = encv[idx];
    }
    __syncthreads();

    v8f acc[NT_TILES] = {};

    const int row   = lane & 15;                        // M row / N col within tile
    const int khalf = lane >> 4;                        // 0 or 1 (K pair select)
    const float2* As2  = reinterpret_cast<const float2*>(As);
    const float2* wt2v = reinterpret_cast<const float2*>(wt2);

    for (int kk = 0; kk < ENC_DIM; kk += 4) {
        const int kpair = (kk >> 1) + khalf;            // index of (k, k+1) pair
        float2 av = As2[row * (ENC_DIM / 2) + kpair];
        v2f a; a.x = av.x; a.y = av.y;

        const float2* brow = wt2v + kpair * NPAD;
        #pragma unroll
        for (int nt = 0; nt < NT_TILES; ++nt) {
            float2 bv = brow[nt * 16 + row];
            v2f b; b.x = bv.x; b.y = bv.y;
            acc[nt] = __builtin_amdgcn_wmma_f32_16x16x4_f32(
                false, a, false, b, (short)0, acc[nt], false, false);
        }
    }

    // Store D + bias; VGPR r -> rows r / r+8 per lane half.
    const int srow = khalf * 8;
    #pragma unroll
    for (int nt = 0; nt < NT_TILES; ++nt) {
        int n = nt * 16 + row;
        if (n < GATES) {
            float bb = bsum[n];
            #pragma unroll
            for (int r = 0; r < 8; ++r) {
                G[(size_t)(mbase + srow + r) * GATES + n] = acc[nt][r] + bb;
            }
        }
    }
}

// ---------------------------------------------------------------------------
// Sequential decode: the 16384 x 5 dependent chain. Single wave32 workgroup
// (barriers in a single-wave WG lower to S_NOP). Everything hot lives in LDS.
// ---------------------------------------------------------------------------
__global__ __launch_bounds__(32) void sd_decode(const float* __restrict__ G,
                                                const float* __restrict__ Tav,
                                                const float* __restrict__ Tpv,
                                                const float* __restrict__ Wlin,
                                                const float* __restrict__ blin,
                                                float* __restrict__ outLogits,
                                                float* __restrict__ outPreds) {
    __shared__ float sTav[N_LABELS * GATES];
    __shared__ float sTpv[N_LABELS * GATES];
    __shared__ float sWlin[N_LABELS * EMB_DIM];
    __shared__ float sBlin[N_LABELS];
    __shared__ float aContrib[GATES];
    __shared__ float gbuf[GATES];
    __shared__ float hbuf[EMB_DIM];
    __shared__ float lbuf[N_LABELS];
    __shared__ int   sPred;
    __shared__ int   p5[MAX_LEN];

    const int lane = threadIdx.x;

    for (int i = lane; i < N_LABELS * GATES; i += 32) { sTav[i] = Tav[i]; sTpv[i] = Tpv[i]; }
    for (int i = lane; i < N_LABELS * EMB_DIM; i += 32) sWlin[i] = Wlin[i];
    if (lane < N_LABELS) sBlin[lane] = blin[lane];
    // t == 0: avg_emb is exactly label_emb[<sow>], not a mean of 6.
    for (int i = lane; i < GATES; i += 32) aContrib[i] = sTav[i];
    __syncthreads();

    for (int t = 0; t < SEQ_LEN; ++t) {
        const float* Grow = G + (size_t)t * GATES;
        if (t + 1 < SEQ_LEN)                                // hide next row's latency
            __builtin_prefetch(G + (size_t)(t + 1) * GATES, 0, 0);

        float gcache[7];
        #pragma unroll
        for (int j = 0; j < 7; ++j) {
            int k = lane + 32 * j;
            gcache[j] = (k < GATES) ? Grow[k] : 0.0f;
        }

        int prev = 0;                                       // <sow> at each position
        for (int s = 0; s < MAX_LEN; ++s) {
            #pragma unroll
            for (int j = 0; j < 7; ++j) {
                int k = lane + 32 * j;
                if (k < GATES)
                    gbuf[k] = gcache[j] + aContrib[k] + sTpv[prev * GATES + k];
            }
            __syncthreads();

            #pragma unroll
            for (int j = 0; j < 2; ++j) {
                int k = lane + 32 * j;
                if (k < EMB_DIM) {
                    // PyTorch gate order (i, f, g, o); c0 = 0 so f-gate is dead.
                    float ig = gbuf[k];
                    float gg = gbuf[100 + k];
                    float og = gbuf[150 + k];
                    float c  = sigmoidf_fast(ig) * tanhf(gg);
                    hbuf[k]  = sigmoidf_fast(og) * tanhf(c);
                }
            }
            __syncthreads();

            if (lane < N_LABELS) {
                float accv = sBlin[lane];
                #pragma unroll 10
                for (int k = 0; k < EMB_DIM; ++k)
                    accv += hbuf[k] * sWlin[lane * EMB_DIM + k];
                lbuf[lane] = accv;
                outLogits[(size_t)(t * MAX_LEN + s) * N_LABELS + lane] = accv;
            }
            __syncthreads();

            if (lane == 0) {
                int best = 0; float bv = lbuf[0];
                for (int j = 1; j < N_LABELS; ++j)          // strict >: first-max, argmax semantics
                    if (lbuf[j] > bv) { bv = lbuf[j]; best = j; }
                sPred = best;
                p5[s] = best;
                outPreds[t * MAX_LEN + s] = (float)best;
            }
            __syncthreads();
            prev = sPred;
        }

        // avg contribution for position t+1: mean over {<sow>} U preds.
        #pragma unroll
        for (int j = 0; j < 7; ++j) {
            int k = lane + 32 * j;
            if (k < GATES) {
                float ssum = sTav[k];                       // index 0 = <sow>
                #pragma unroll
                for (int s = 0; s < MAX_LEN; ++s)
                    ssum += sTav[p5[s] * GATES + k];
                aContrib[k] = ssum * (1.0f / 6.0f);
            }
        }
        __syncthreads();
    }
}

// ---------------------------------------------------------------------------
// Launch
// ---------------------------------------------------------------------------
extern "C" void kernel_launch(void* const* d_in, const int* in_sizes, int n_in,
                              void* d_out, int out_size, void* d_ws, size_t ws_size,
                              hipStream_t stream) {
    const float* enc   = (const float*)d_in[0];   // [1, 16384, 768]
    const float* emb   = (const float*)d_in[1];   // [9, 50]
    const float* Wih   = (const float*)d_in[2];   // [200, 868]
    // d_in[3] = W_hh (unused: h0 = 0)
    const float* bih   = (const float*)d_in[4];   // [200]
    const float* bhh   = (const float*)d_in[5];   // [200]
    const float* Wlin  = (const float*)d_in[6];   // [9, 50]
    const float* blin  = (const float*)d_in[7];   // [9]

    float* out       = (float*)d_out;
    float* outLogits = out;                               // 16384*5*9
    float* outPreds  = out + (size_t)SEQ_LEN * MAX_LEN * N_LABELS;

    // Workspace layout (floats)
    float* ws   = (float*)d_ws;
    float* G    = ws;                                     // 16384*200   = 3,276,800
    float* wt2  = G    + (size_t)SEQ_LEN * GATES;         // 384*208*2   =   159,744
    float* bsum = wt2  + (size_t)(ENC_DIM / 2) * NPAD * 2;//               200
    float* Tav  = bsum + 256;                             // 9*200       =     1,800
    float* Tpv  = Tav  + N_LABELS * GATES;                // 9*200       =     1,800

    // 1) Repack weights + build tables.
    {
        int n = (ENC_DIM / 2) * NPAD;
        sd_prep_wt2<<<(n + 255) / 256, 256, 0, stream>>>(Wih, wt2);
        sd_prep_tables<<<1, 256, 0, stream>>>(Wih, emb, bih, bhh, bsum, Tav, Tpv);
    }
    // 2) WMMA GEMM for the encoder contribution to all gates.
    sd_gemm_gates<<<SEQ_LEN / 16, 32, 0, stream>>>(enc, wt2, bsum, G);
    // 3) Latency-bound sequential decode (single wave32).
    sd_decode<<<1, 32, 0, stream>>>(G, Tav, Tpv, Wlin, blin, outLogits, outPreds);
}